// HandModel_58669253263387
// MI455X (gfx1250) — compile-verified
//
#include <hip/hip_runtime.h>
#include <math.h>

// ---------- problem constants ----------
constexpr int NB    = 8;       // batch
constexpr int NV    = 778;     // vertices
constexpr int NVP   = 784;     // padded to 49*16
constexpr int NJT   = 16;      // joints
constexpr int NCOMP = 24;
constexpr int NBET  = 10;
constexpr int NF    = 1554;    // faces
constexpr int NANCH = 32;
constexpr int NO    = 16384;   // object points

// ---------- output layout (floats) ----------
constexpr int OFF_CMAP  = 0;                         // NB*NO
constexpr int OFF_PEN   = NB * NO;                   // 131072 (1 float)
constexpr int OFF_VERTS = OFF_PEN + 1;               // 131073 (NB*NV*3)
constexpr int OFF_CC    = OFF_VERTS + NB * NV * 3;   // 149745 (NB*NANCH*3)

// ---------- workspace layout (floats) ----------
constexpr int WS_VSHAPED = 0;                      // NB*NV*3 = 18672
constexpr int WS_VPOSED  = WS_VSHAPED + NB*NV*3;   // 18672
constexpr int WS_R       = WS_VPOSED  + NB*NV*3;   // NB*NJT*9 = 1152
constexpr int WS_J       = WS_R       + NB*NJT*9;  // NB*NJT*3 = 384
constexpr int WS_PF      = WS_J       + NB*NJT*3;  // NB*135 = 1080
constexpr int WS_T       = WS_PF      + NB*135;    // NB*NJT*16 = 2048
constexpr int WS_FULL    = WS_T       + NB*NJT*16; // NB*48 = 384
constexpr int WS_NRM     = WS_FULL    + NB*48;     // NB*NV*3 = 18672

typedef __attribute__((ext_vector_type(2))) float v2f;
typedef __attribute__((ext_vector_type(8))) float v8f;

// =====================================================================
// Kernel 1: full MANO pipeline in one block (tiny serial-ish work)
// =====================================================================
__global__ __launch_bounds__(256)
void mano_kernel(const float* __restrict__ pose, const float* __restrict__ beta,
                 const float* __restrict__ v_template, const float* __restrict__ shapedirs,
                 const float* __restrict__ posedirs, const float* __restrict__ Jreg,
                 const float* __restrict__ lbs, const float* __restrict__ pca,
                 const float* __restrict__ anchor_w, const int* __restrict__ anchor_idx,
                 float* __restrict__ out, float* __restrict__ ws)
{
    const int tid = threadIdx.x;
    float* vshaped = ws + WS_VSHAPED;
    float* vposed  = ws + WS_VPOSED;
    float* Rm      = ws + WS_R;
    float* Jm      = ws + WS_J;
    float* pf      = ws + WS_PF;
    float* Tm      = ws + WS_T;
    float* full    = ws + WS_FULL;
    float* vnrm    = ws + WS_NRM;

    // zero normals accumulator + penetration (d_out/d_ws are poisoned)
    for (int i = tid; i < NB*NV*3; i += 256) vnrm[i] = 0.f;
    if (tid == 0) out[OFF_PEN] = 0.f;

    // full pose: [root(3) | pca@comps(45)] per batch
    for (int i = tid; i < NB*48; i += 256) {
        int b = i / 48, c = i % 48;
        float v;
        if (c < 3) v = pose[b*30 + 3 + c];
        else {
            int p = c - 3; v = 0.f;
            for (int k = 0; k < NCOMP; ++k) v = fmaf(pose[b*30 + 6 + k], pca[k*45 + p], v);
        }
        full[i] = v;
    }
    __syncthreads();

    // Rodrigues: 128 rotation matrices
    for (int i = tid; i < NB*NJT; i += 256) {
        float rx = full[i*3+0], ry = full[i*3+1], rz = full[i*3+2];
        float th = sqrtf(rx*rx + ry*ry + rz*rz + 1e-16f);
        float inv = 1.f / th;
        float kx = rx*inv, ky = ry*inv, kz = rz*inv;
        float c = cosf(th), s = sinf(th), o = 1.f - c;
        float* R = Rm + i*9;
        R[0]=c+o*kx*kx;     R[1]=-s*kz+o*kx*ky; R[2]= s*ky+o*kx*kz;
        R[3]= s*kz+o*ky*kx; R[4]=c+o*ky*ky;     R[5]=-s*kx+o*ky*kz;
        R[6]=-s*ky+o*kz*kx; R[7]= s*kx+o*kz*ky; R[8]=c+o*kz*kz;
    }

    // v_shaped = template + shapedirs . beta
    for (int i = tid; i < NB*NV*3; i += 256) {
        int b = i/(NV*3), r = i%(NV*3), v = r/3, d = r%3;
        float acc = v_template[v*3 + d];
        for (int k = 0; k < NBET; ++k)
            acc = fmaf(shapedirs[(v*3+d)*NBET + k], beta[b*NBET + k], acc);
        vshaped[i] = acc;
    }
    __syncthreads();

    // J = Jreg @ v_shaped ; pose_feat = R[1:] - I
    for (int i = tid; i < NB*NJT*3; i += 256) {
        int b = i/(NJT*3), r = i%(NJT*3), j = r/3, d = r%3;
        float acc = 0.f;
        const float* vs = vshaped + b*NV*3 + d;
        const float* jr = Jreg + j*NV;
        for (int v = 0; v < NV; ++v) acc = fmaf(jr[v], vs[v*3], acc);
        Jm[i] = acc;
    }
    for (int i = tid; i < NB*135; i += 256) {
        int b = i/135, p = i%135, j = p/9 + 1, e = p%9;
        pf[i] = Rm[(b*NJT+j)*9 + e] - ((e==0 || e==4 || e==8) ? 1.f : 0.f);
    }
    __syncthreads();

    // v_posed = v_shaped + posedirs . pose_feat
    for (int i = tid; i < NB*NV*3; i += 256) {
        int b = i/(NV*3), r = i%(NV*3), v = r/3, d = r%3;
        float acc = vshaped[i];
        const float* pd = posedirs + (v*3+d)*135;
        const float* f  = pf + b*135;
        for (int p = 0; p < 135; ++p) acc = fmaf(pd[p], f[p], acc);
        vposed[i] = acc;
    }

    // kinematic chain, level by level (depth = 0..3)
    for (int level = 0; level < 4; ++level) {
        __syncthreads();
        for (int i = tid; i < NB*NJT; i += 256) {
            int b = i/NJT, j = i%NJT;
            int depth = (j == 0) ? 0 : (1 + (j-1)%3);
            if (depth != level) continue;
            int par = (j == 0) ? -1 : (((j-1)%3 == 0) ? 0 : j-1);
            const float* R = Rm + i*9;
            float t0, t1, t2;
            if (j == 0) { t0 = Jm[b*48+0]; t1 = Jm[b*48+1]; t2 = Jm[b*48+2]; }
            else {
                t0 = Jm[b*48 + j*3+0] - Jm[b*48 + par*3+0];
                t1 = Jm[b*48 + j*3+1] - Jm[b*48 + par*3+1];
                t2 = Jm[b*48 + j*3+2] - Jm[b*48 + par*3+2];
            }
            float L[16] = { R[0],R[1],R[2],t0,  R[3],R[4],R[5],t1,
                            R[6],R[7],R[8],t2,  0.f,0.f,0.f,1.f };
            float* T = Tm + i*16;
            if (j == 0) {
                for (int e = 0; e < 16; ++e) T[e] = L[e];
            } else {
                const float* P = Tm + (b*NJT + par)*16;
                float tmp[16];
                for (int mm = 0; mm < 4; ++mm)
                    for (int nn = 0; nn < 4; ++nn) {
                        float s = 0.f;
                        for (int kk = 0; kk < 4; ++kk) s = fmaf(P[mm*4+kk], L[kk*4+nn], s);
                        tmp[mm*4+nn] = s;
                    }
                for (int e = 0; e < 16; ++e) T[e] = tmp[e];
            }
        }
    }
    __syncthreads();

    // A = T with translation corrected: t -= Rglobal @ J
    for (int i = tid; i < NB*NJT; i += 256) {
        int b = i/NJT, j = i%NJT;
        float* T = Tm + i*16;
        float jx = Jm[b*48+j*3+0], jy = Jm[b*48+j*3+1], jz = Jm[b*48+j*3+2];
        float c0 = T[0]*jx + T[1]*jy + T[2]*jz;
        float c1 = T[4]*jx + T[5]*jy + T[6]*jz;
        float c2 = T[8]*jx + T[9]*jy + T[10]*jz;
        T[3] -= c0; T[7] -= c1; T[11] -= c2;
    }
    __syncthreads();

    // LBS: verts = (sum_j w * A) applied to v_posed, plus trans
    for (int i = tid; i < NB*NV; i += 256) {
        int b = i/NV, v = i%NV;
        float Tw[12];
        for (int e = 0; e < 12; ++e) Tw[e] = 0.f;
        for (int j = 0; j < NJT; ++j) {
            float w = lbs[v*NJT + j];
            const float* A = Tm + (b*NJT + j)*16;
            for (int e = 0; e < 12; ++e) Tw[e] = fmaf(w, A[e], Tw[e]);
        }
        float px = vposed[b*NV*3 + v*3+0];
        float py = vposed[b*NV*3 + v*3+1];
        float pz = vposed[b*NV*3 + v*3+2];
        float ox = Tw[0]*px + Tw[1]*py + Tw[2]*pz  + Tw[3]  + pose[b*30+0];
        float oy = Tw[4]*px + Tw[5]*py + Tw[6]*pz  + Tw[7]  + pose[b*30+1];
        float oz = Tw[8]*px + Tw[9]*py + Tw[10]*pz + Tw[11] + pose[b*30+2];
        out[OFF_VERTS + b*NV*3 + v*3+0] = ox;
        out[OFF_VERTS + b*NV*3 + v*3+1] = oy;
        out[OFF_VERTS + b*NV*3 + v*3+2] = oz;
    }
    __syncthreads();

    // contact candidates
    for (int i = tid; i < NB*NANCH*3; i += 256) {
        int b = i/(NANCH*3), r = i%(NANCH*3), a = r/3, d = r%3;
        float acc = 0.f;
        for (int k = 0; k < 3; ++k) {
            int vi = anchor_idx[a*3 + k];
            acc = fmaf(anchor_w[a*3 + k], out[OFF_VERTS + b*NV*3 + vi*3 + d], acc);
        }
        out[OFF_CC + i] = acc;
    }
}

// =====================================================================
// Kernel 2: face normals -> scatter-add into workspace
// =====================================================================
__global__ __launch_bounds__(256)
void fnorm_kernel(const int* __restrict__ faces, const float* __restrict__ verts_all,
                  float* __restrict__ vnrm)
{
    int i = blockIdx.x * 256 + threadIdx.x;
    if (i >= NB*NF) return;
    int b = i / NF, f = i % NF;
    const float* verts = verts_all + b*NV*3;
    int i0 = faces[f*3+0], i1 = faces[f*3+1], i2 = faces[f*3+2];
    float p0x = verts[i0*3], p0y = verts[i0*3+1], p0z = verts[i0*3+2];
    float e1x = verts[i1*3]   - p0x, e1y = verts[i1*3+1] - p0y, e1z = verts[i1*3+2] - p0z;
    float e2x = verts[i2*3]   - p0x, e2y = verts[i2*3+1] - p0y, e2z = verts[i2*3+2] - p0z;
    float fnx = e1y*e2z - e1z*e2y;
    float fny = e1z*e2x - e1x*e2z;
    float fnz = e1x*e2y - e1y*e2x;
    float* base = vnrm + b*NV*3;
    atomicAdd(base + i0*3+0, fnx); atomicAdd(base + i0*3+1, fny); atomicAdd(base + i0*3+2, fnz);
    atomicAdd(base + i1*3+0, fnx); atomicAdd(base + i1*3+1, fny); atomicAdd(base + i1*3+2, fnz);
    atomicAdd(base + i2*3+0, fnx); atomicAdd(base + i2*3+1, fny); atomicAdd(base + i2*3+2, fnz);
}

// =====================================================================
// Kernel 3: normalize vertex normals in place
// =====================================================================
__global__ __launch_bounds__(256)
void nrm_kernel(float* __restrict__ vnrm)
{
    int i = blockIdx.x * 256 + threadIdx.x;
    if (i >= NB*NV) return;
    float x = vnrm[i*3], y = vnrm[i*3+1], z = vnrm[i*3+2];
    float inv = 1.f / (sqrtf(x*x + y*y + z*z) + 1e-8f);
    vnrm[i*3] = x*inv; vnrm[i*3+1] = y*inv; vnrm[i*3+2] = z*inv;
}

// =====================================================================
// Kernel 4: nearest-neighbor via V_WMMA_F32_16X16X4_F32
//   32 points per wave: two A fragments (16 pts each) share one B fragment
//   loaded as a single ds_load_b128 (x,y,z,|v|^2) per lane per tile.
//   A layout: lanes 0-15 hold (x,y)=K0,K1; lanes 16-31 hold (z,0)=K2,K3.
//   D layout: lane L holds 8 slots, M = r + 8*(L>=16), N = L&15.
//   argmin key = |v|^2 - 2*dot; |o|^2 added once at the end.
// =====================================================================
__global__ __launch_bounds__(256)
void nn_kernel(const float* __restrict__ obj, const float* __restrict__ verts_all,
               const float* __restrict__ vnrm, float* __restrict__ out)
{
    __shared__ float4 s_v[NVP];
    const int b   = blockIdx.y;
    const int tid = threadIdx.x;
    const float* verts = verts_all + b*NV*3;

    for (int i = tid; i < NVP; i += 256) {
        float x = 0.f, y = 0.f, z = 0.f, n2 = 1e30f;   // pad verts never win argmin
        if (i < NV) { x = verts[i*3]; y = verts[i*3+1]; z = verts[i*3+2]; n2 = x*x+y*y+z*z; }
        s_v[i] = make_float4(x, y, z, n2);
    }
    __syncthreads();

    const int lane = tid & 31, wave = tid >> 5;
    const int m0 = blockIdx.x * 256 + wave * 32;       // 32 points per wave
    const int m  = lane & 15;
    const bool hi = lane >= 16;

    // lane m (and lane 16+m) load coords of points m and 16+m of this chunk
    const float* pA = obj + (size_t)(b*NO + m0 + m) * 3;
    const float* pB = obj + (size_t)(b*NO + m0 + 16 + m) * 3;
    float axA = pA[0], ayA = pA[1], azA = pA[2];
    float axB = pB[0], ayB = pB[1], azB = pB[2];

    v2f a0, a1;
    a0.x = hi ? azA : axA;  a0.y = hi ? 0.f : ayA;
    a1.x = hi ? azB : axB;  a1.y = hi ? 0.f : ayB;

    float mind0[8], mind1[8]; int mini0[8], mini1[8];
    #pragma unroll
    for (int r = 0; r < 8; ++r) { mind0[r] = 3.4e38f; mind1[r] = 3.4e38f; mini0[r] = 0; mini1[r] = 0; }

    for (int t = 0; t < NVP/16; ++t) {
        int vidx = t*16 + m;
        float4 vv = s_v[vidx];                       // one ds_load_b128, no predication
        v2f bv; bv.x = hi ? vv.z : vv.x;  bv.y = hi ? 0.f : vv.y;
        v8f c0 = {}, c1 = {};
        c0 = __builtin_amdgcn_wmma_f32_16x16x4_f32(false, a0, false, bv, (short)0, c0, false, false);
        c1 = __builtin_amdgcn_wmma_f32_16x16x4_f32(false, a1, false, bv, (short)0, c1, false, false);
        #pragma unroll
        for (int r = 0; r < 8; ++r) {
            float k0 = fmaf(-2.f, c0[r], vv.w);
            float k1 = fmaf(-2.f, c1[r], vv.w);
            if (k0 < mind0[r]) { mind0[r] = k0; mini0[r] = vidx; }
            if (k1 < mind1[r]) { mind1[r] = k1; mini1[r] = vidx; }
        }
    }

    // reduce (key,idx) across the 16 lanes of each half (xor<16 stays in half)
    #pragma unroll
    for (int mask = 1; mask < 16; mask <<= 1) {
        #pragma unroll
        for (int r = 0; r < 8; ++r) {
            float od = __shfl_xor(mind0[r], mask, 32);
            int   oi = __shfl_xor(mini0[r], mask, 32);
            if (od < mind0[r] || (od == mind0[r] && oi < mini0[r])) { mind0[r] = od; mini0[r] = oi; }
            od = __shfl_xor(mind1[r], mask, 32);
            oi = __shfl_xor(mini1[r], mask, 32);
            if (od < mind1[r] || (od == mind1[r] && oi < mini1[r])) { mind1[r] = od; mini1[r] = oi; }
        }
    }

    // hand each point's result to lane m: point m <- A0 slot, point 16+m <- A1 slot
    float k0 = 0.f, k1 = 0.f; int i0 = 0, i1 = 0;
    #pragma unroll
    for (int r = 0; r < 8; ++r) {
        float dl = __shfl(mind0[r], 0, 32);  int il = __shfl(mini0[r], 0, 32);
        float du = __shfl(mind0[r], 16, 32); int iu = __shfl(mini0[r], 16, 32);
        if (lane == r)     { k0 = dl; i0 = il; }
        if (lane == r + 8) { k0 = du; i0 = iu; }
        dl = __shfl(mind1[r], 0, 32);  il = __shfl(mini1[r], 0, 32);
        du = __shfl(mind1[r], 16, 32); iu = __shfl(mini1[r], 16, 32);
        if (lane == r)     { k1 = dl; i1 = il; }
        if (lane == r + 8) { k1 = du; i1 = iu; }
    }

    float pen = 0.f;
    if (!hi) {
        float onA = axA*axA + ayA*ayA + azA*azA;
        float onB = axB*axB + ayB*ayB + azB*azB;
        float d0 = k0 + onA, d1 = k1 + onB;          // full d2 = |o|^2 + key
        // cmap = 1 - 2*(sigmoid(100 d) - 0.5) = 2 / (1 + exp(100 d))
        out[OFF_CMAP + b*NO + m0 + m]      = 2.f / (1.f + __expf(100.f * d0));
        out[OFF_CMAP + b*NO + m0 + 16 + m] = 2.f / (1.f + __expf(100.f * d1));
        const float* nv0 = verts + i0*3;
        const float* nn0 = vnrm + ((size_t)b*NV + i0)*3;
        float dp0 = (nv0[0]-axA)*nn0[0] + (nv0[1]-ayA)*nn0[1] + (nv0[2]-azA)*nn0[2];
        if (dp0 > 0.f) pen += d0;
        const float* nv1 = verts + i1*3;
        const float* nn1 = vnrm + ((size_t)b*NV + i1)*3;
        float dp1 = (nv1[0]-axB)*nn1[0] + (nv1[1]-ayB)*nn1[1] + (nv1[2]-azB)*nn1[2];
        if (dp1 > 0.f) pen += d1;
    }
    #pragma unroll
    for (int mask = 16; mask >= 1; mask >>= 1) pen += __shfl_xor(pen, mask, 32);
    if (lane == 0) atomicAdd(out + OFF_PEN, pen * (1.0f / NB));
}

// =====================================================================
extern "C" void kernel_launch(void* const* d_in, const int* in_sizes, int n_in,
                              void* d_out, int out_size, void* d_ws, size_t ws_size,
                              hipStream_t stream)
{
    const float* hand_pose      = (const float*)d_in[0];
    const float* obj_points     = (const float*)d_in[1];
    const float* hand_beta      = (const float*)d_in[2];
    const float* v_template     = (const float*)d_in[3];
    const float* shapedirs      = (const float*)d_in[4];
    const float* posedirs       = (const float*)d_in[5];
    const float* J_regressor    = (const float*)d_in[6];
    const float* lbs_weights    = (const float*)d_in[7];
    const float* pca_comps      = (const float*)d_in[8];
    const float* anchor_weights = (const float*)d_in[9];
    const int*   closed_faces   = (const int*)d_in[10];
    const int*   anchor_face_idx= (const int*)d_in[11];
    float* out = (float*)d_out;
    float* ws  = (float*)d_ws;
    float* vnrm = ws + WS_NRM;

    mano_kernel<<<1, 256, 0, stream>>>(hand_pose, hand_beta, v_template, shapedirs,
                                       posedirs, J_regressor, lbs_weights, pca_comps,
                                       anchor_weights, anchor_face_idx, out, ws);
    fnorm_kernel<<<(NB*NF + 255)/256, 256, 0, stream>>>(closed_faces, out + OFF_VERTS, vnrm);
    nrm_kernel<<<(NB*NV + 255)/256, 256, 0, stream>>>(vnrm);
    dim3 grid(NO/256, NB);
    nn_kernel<<<grid, 256, 0, stream>>>(obj_points, out + OFF_VERTS, vnrm, out);
}